// GCN_35966056137205
// MI455X (gfx1250) — compile-verified
//
#include <hip/hip_runtime.h>

typedef __attribute__((ext_vector_type(2))) float v2f;
typedef __attribute__((ext_vector_type(8))) float v8f;

#define F_IN 512
#define HID  16
#define NCLS 40
#define NCP  48   // N_CLASSES padded to 3 WMMA n-tiles

// ---------------- init / degree ----------------

__global__ void fill_f32(float* __restrict__ p, float v, int n) {
  int i = blockIdx.x * blockDim.x + threadIdx.x;
  int stride = gridDim.x * blockDim.x;
  for (; i < n; i += stride) p[i] = v;
}

__global__ void deg_accum(const int* __restrict__ col, float* __restrict__ deg, int E) {
  int e = blockIdx.x * blockDim.x + threadIdx.x;
  if (e < E) atomicAdd(&deg[col[e]], 1.0f);
}

__global__ void deg_to_rsqrt(float* __restrict__ d, int n) {
  int i = blockIdx.x * blockDim.x + threadIdx.x;
  if (i < n) d[i] = rsqrtf(d[i]);   // deg >= 1 always (self loop)
}

// ---------------- layer 1: h1s = (x @ W1) * dinv[row], fp32 WMMA ----------------
// One wave per 16-row tile; A fragment: lane l -> M=l%16, K=2*(l/16)+{0,1}
// B fragment (from LDS): lane l -> N=l%16, K=2*(l/16)+{0,1}
// C/D: VGPR r -> M = 8*(l/16)+r, N = l%16

__global__ void __launch_bounds__(128) gemm1_wmma(
    const float* __restrict__ x, const float* __restrict__ W1,
    const float* __restrict__ dinv, float* __restrict__ h1s, int ntiles) {
  __shared__ float w1s[F_IN * HID];
  for (int i = threadIdx.x; i < F_IN * HID; i += 128) w1s[i] = W1[i];
  __syncthreads();

  const int wave = threadIdx.x >> 5;
  const int lane = threadIdx.x & 31;
  const int tile = blockIdx.x * 4 + wave;
  if (tile >= ntiles) return;

  const int m  = lane & 15;          // A-row / B-col within tile
  const int kh = (lane >> 4) << 1;   // 0 or 2: K sub-offset for upper half-wave
  const float* xrow = x + (size_t)(tile * 16 + m) * F_IN;

  v8f c = {};
#pragma unroll 8
  for (int k0 = 0; k0 < F_IN; k0 += 4) {
    const int k = k0 + kh;
    v2f a, b;
    a.x = xrow[k];
    a.y = xrow[k + 1];
    b.x = w1s[k * HID + m];
    b.y = w1s[(k + 1) * HID + m];
    c = __builtin_amdgcn_wmma_f32_16x16x4_f32(false, a, false, b, (short)0, c,
                                              false, false);
  }

  const int mbase = tile * 16 + ((lane >> 4) << 3);
#pragma unroll
  for (int r = 0; r < 8; ++r) {
    const int rr = mbase + r;
    h1s[(size_t)rr * HID + m] = c[r] * dinv[rr];
  }
}

// ---------------- edge scatter, layer 1 (16 features/edge) ----------------

__global__ void scatter1(const int* __restrict__ row, const int* __restrict__ col,
                         const float* __restrict__ h1s, float* __restrict__ agg1,
                         int E) {
  long long t = (long long)blockIdx.x * blockDim.x + threadIdx.x;
  if (t >= (long long)E * HID) return;
  const int e = (int)(t >> 4);
  const int f = (int)(t & 15);
  atomicAdd(&agg1[(size_t)col[e] * HID + f], h1s[(size_t)row[e] * HID + f]);
}

// agg1 <- relu( (agg1 + h1s_selfloop) * dinv + b1 ), in place
__global__ void finish1(float* __restrict__ agg, const float* __restrict__ h1s,
                        const float* __restrict__ dinv,
                        const float* __restrict__ b1, int n) {
  int i = blockIdx.x * blockDim.x + threadIdx.x;
  if (i >= n) return;
  const int node = i >> 4;
  const int f = i & 15;
  float v = (agg[i] + h1s[i]) * dinv[node] + b1[f];
  agg[i] = v > 0.0f ? v : 0.0f;
}

// ---------------- layer 2: h2s = (h @ W2) * dinv[row], fp32 WMMA, N padded ----------------

__global__ void __launch_bounds__(128) gemm2_wmma(
    const float* __restrict__ h, const float* __restrict__ W2,
    const float* __restrict__ dinv, float* __restrict__ h2s, int ntiles) {
  __shared__ float w2s[HID * NCP];
  for (int i = threadIdx.x; i < HID * NCP; i += 128) {
    const int k = i / NCP, n = i % NCP;
    w2s[i] = (n < NCLS) ? W2[k * NCLS + n] : 0.0f;
  }
  __syncthreads();

  const int wave = threadIdx.x >> 5;
  const int lane = threadIdx.x & 31;
  const int tile = blockIdx.x * 4 + wave;
  if (tile >= ntiles) return;

  const int m  = lane & 15;
  const int kh = (lane >> 4) << 1;
  const float* hrow = h + (size_t)(tile * 16 + m) * HID;

  float areg[8];
#pragma unroll
  for (int k0 = 0, j = 0; k0 < HID; k0 += 4, j += 2) {
    areg[j]     = hrow[k0 + kh];
    areg[j + 1] = hrow[k0 + kh + 1];
  }
  const int mbase = tile * 16 + ((lane >> 4) << 3);
  float dreg[8];
#pragma unroll
  for (int r = 0; r < 8; ++r) dreg[r] = dinv[mbase + r];

#pragma unroll
  for (int nt = 0; nt < 3; ++nt) {
    v8f c = {};
#pragma unroll
    for (int k0 = 0, j = 0; k0 < HID; k0 += 4, j += 2) {
      const int k = k0 + kh;
      v2f a, b;
      a.x = areg[j];
      a.y = areg[j + 1];
      b.x = w2s[k * NCP + nt * 16 + m];
      b.y = w2s[(k + 1) * NCP + nt * 16 + m];
      c = __builtin_amdgcn_wmma_f32_16x16x4_f32(false, a, false, b, (short)0, c,
                                                false, false);
    }
    const int n = nt * 16 + m;
    if (n < NCLS) {
#pragma unroll
      for (int r = 0; r < 8; ++r) {
        h2s[(size_t)(mbase + r) * NCLS + n] = c[r] * dreg[r];
      }
    }
  }
}

// ---------------- edge scatter, layer 2 (40 features/edge) ----------------

__global__ void scatter2(const int* __restrict__ row, const int* __restrict__ col,
                         const float* __restrict__ h2s, float* __restrict__ out,
                         int E) {
  long long t = (long long)blockIdx.x * blockDim.x + threadIdx.x;
  if (t >= (long long)E * NCLS) return;
  const int e = (int)(t / NCLS);
  const int f = (int)(t - (long long)e * NCLS);
  atomicAdd(&out[(size_t)col[e] * NCLS + f], h2s[(size_t)row[e] * NCLS + f]);
}

// out <- (out + h2s_selfloop) * dinv + b2, in place
__global__ void finish2(float* __restrict__ out, const float* __restrict__ h2s,
                        const float* __restrict__ dinv,
                        const float* __restrict__ b2, int n) {
  int i = blockIdx.x * blockDim.x + threadIdx.x;
  if (i >= n) return;
  const int node = i / NCLS;
  const int f = i - node * NCLS;
  out[i] = (out[i] + h2s[i]) * dinv[node] + b2[f];
}

// ---------------- host ----------------

extern "C" void kernel_launch(void* const* d_in, const int* in_sizes, int n_in,
                              void* d_out, int out_size, void* d_ws, size_t ws_size,
                              hipStream_t stream) {
  const float* x  = (const float*)d_in[0];
  const int*   ei = (const int*)d_in[1];   // [2, E] int32
  const float* W1 = (const float*)d_in[2];
  const float* b1 = (const float*)d_in[3];
  const float* W2 = (const float*)d_in[4];
  const float* b2 = (const float*)d_in[5];

  const int N = in_sizes[0] / F_IN;
  const int E = in_sizes[1] / 2;
  const int* row = ei;
  const int* col = ei + E;

  // workspace layout (floats): dinv[N] | h1s[N*16] | agg1[N*16] | h2s[N*40]
  float* ws   = (float*)d_ws;
  float* dinv = ws;
  float* h1s  = dinv + N;
  float* agg1 = h1s + (size_t)N * HID;
  float* h2s  = agg1 + (size_t)N * HID;
  float* out  = (float*)d_out;   // doubles as layer-2 scatter accumulator

  // init: deg starts at 1.0 (self loop), accumulators at 0
  fill_f32<<<1024, 256, 0, stream>>>(dinv, 1.0f, N);
  fill_f32<<<2048, 256, 0, stream>>>(agg1, 0.0f, N * HID);
  fill_f32<<<4096, 256, 0, stream>>>(out, 0.0f, N * NCLS);

  deg_accum<<<(E + 255) / 256, 256, 0, stream>>>(col, dinv, E);
  deg_to_rsqrt<<<(N + 255) / 256, 256, 0, stream>>>(dinv, N);

  const int ntiles = (N + 15) / 16;        // 6250, exact
  const int gtiles = (ntiles + 3) / 4;     // 4 waves (tiles) per 128-thread block

  gemm1_wmma<<<gtiles, 128, 0, stream>>>(x, W1, dinv, h1s, ntiles);

  const long long t1 = (long long)E * HID;
  scatter1<<<(int)((t1 + 255) / 256), 256, 0, stream>>>(row, col, h1s, agg1, E);
  finish1<<<(N * HID + 255) / 256, 256, 0, stream>>>(agg1, h1s, dinv, b1, N * HID);

  gemm2_wmma<<<gtiles, 128, 0, stream>>>(agg1, W2, dinv, h2s, ntiles);

  const long long t2 = (long long)E * NCLS;
  scatter2<<<(int)((t2 + 255) / 256), 256, 0, stream>>>(row, col, h2s, out, E);
  finish2<<<(N * NCLS + 255) / 256, 256, 0, stream>>>(out, h2s, dinv, b2, N * NCLS);
}